// Sampler_24446953849417
// MI455X (gfx1250) — compile-verified
//
#include <hip/hip_runtime.h>
#include <hip/hip_bf16.h>
#include <stdint.h>

#define THREADS     256
#define TILE_FLOATS 1024          // THREADS * 4 floats per async tile (4 KB)
#define NBINS       2048          // 11-bit radix histogram
#define NSUB        4             // privatized histogram copies (interleaved)
#define CAND_N      2048          // candidate buffer (64-bit composite keys)
#define TEMP_INV    1.25f         // 1 / 0.8
#define TOPP        0.9f

typedef float v2f __attribute__((ext_vector_type(2)));
typedef float v8f __attribute__((ext_vector_type(8)));

// Order-preserving float -> uint key (larger float => larger key)
__device__ __forceinline__ unsigned fkey(float f) {
  unsigned u = __float_as_uint(f);
  return (u & 0x80000000u) ? ~u : (u | 0x80000000u);
}
__device__ __forceinline__ float keyval(unsigned k) {
  unsigned u = (k & 0x80000000u) ? (k & 0x7FFFFFFFu) : ~k;
  return __uint_as_float(u);
}

__global__ __launch_bounds__(THREADS)
void sampler_topk_topp_kernel(const float* __restrict__ logits,
                              const float* __restrict__ u,
                              const int* __restrict__ topk_p,
                              int* __restrict__ out,
                              int B, int V)
{
  __shared__ float              stage[2][TILE_FLOATS];   // async staging (8 KB)
  __shared__ unsigned           histp[NBINS * NSUB];     // 32 KB, copies interleaved
  __shared__ unsigned           chunk[THREADS];          // suffix-scan scratch
  __shared__ unsigned long long cand[CAND_N];            // 16 KB composite keys
  __shared__ float              wbuf[256];               // exp-weights (zero-padded)
  __shared__ int                s_fidx[64];              // finalist vocab indices
  __shared__ float              scanbuf[64];             // WMMA scan result
  __shared__ unsigned           s_thresh, s_ccount;
  __shared__ int                s_K;

  const int tid = threadIdx.x;
  const int row = blockIdx.x;
  const float* rowp = logits + (size_t)row * (size_t)V;

  // ---- init shared state ----
  for (int i = tid; i < NBINS * NSUB; i += THREADS) histp[i] = 0u;
  for (int i = tid; i < CAND_N; i += THREADS)       cand[i] = 0ull;
  if (tid == 0) { s_ccount = 0u; s_thresh = 0u; s_K = *topk_p; }
  __syncthreads();

  const unsigned sub = (unsigned)(tid & (NSUB - 1)); // spreads lanes of a wave

  // =====================================================================
  // Pass 1: stream row via CDNA5 async global->LDS copies (double-buffered,
  // ASYNCcnt-pipelined), build privatized 11-bit radix histogram.
  // =====================================================================
  const int NT = V / TILE_FLOATS;
  {
    int buf = 0;
    if (NT > 0) {
      unsigned loff = (unsigned)(uintptr_t)&stage[0][tid * 4];
      const float* g = rowp + tid * 4;
      asm volatile("global_load_async_to_lds_b128 %0, %1, off"
                   :: "v"(loff), "v"(g) : "memory");
    }
    for (int t = 0; t < NT; ++t) {
      if (t + 1 < NT) {
        unsigned loff = (unsigned)(uintptr_t)&stage[buf ^ 1][tid * 4];
        const float* g = rowp + (size_t)(t + 1) * TILE_FLOATS + tid * 4;
        asm volatile("global_load_async_to_lds_b128 %0, %1, off"
                     :: "v"(loff), "v"(g) : "memory");
        asm volatile("s_wait_asynccnt 0x1" ::: "memory"); // tile t landed
      } else {
        asm volatile("s_wait_asynccnt 0x0" ::: "memory");
      }
      __syncthreads();                 // all waves' portions visible
      #pragma unroll
      for (int c = 0; c < 4; ++c) {    // stride-1 reads: conflict-free banks
        float f = stage[buf][c * THREADS + tid];
        atomicAdd(&histp[(fkey(f) >> 21) * NSUB + sub], 1u);
      }
      __syncthreads();                 // done reading before buf reuse
      buf ^= 1;
    }
    for (int i = NT * TILE_FLOATS + tid; i < V; i += THREADS)
      atomicAdd(&histp[(fkey(rowp[i]) >> 21) * NSUB + sub], 1u);
  }
  __syncthreads();

  // =====================================================================
  // Radix threshold T: largest bin with suffix-count >= K.
  // =====================================================================
  {
    const int PB = NBINS / THREADS;    // 8 bins per thread
    const int base = tid * PB;
    unsigned s = 0;
    #pragma unroll
    for (int b = 0; b < PB; ++b) {
      #pragma unroll
      for (int c = 0; c < NSUB; ++c) s += histp[(base + b) * NSUB + c];
    }
    chunk[tid] = s;
    __syncthreads();
    for (int off = 1; off < THREADS; off <<= 1) {   // Hillis-Steele suffix scan
      unsigned add = (tid + off < THREADS) ? chunk[tid + off] : 0u;
      __syncthreads();
      chunk[tid] += add;
      __syncthreads();
    }
    int Kc = s_K; Kc = Kc < 1 ? 1 : (Kc > 64 ? 64 : Kc);
    unsigned K = (unsigned)Kc;
    bool cross = (chunk[tid] >= K) &&
                 (tid == THREADS - 1 || chunk[tid + 1] < K);
    if (cross) {
      unsigned acc = (tid == THREADS - 1) ? 0u : chunk[tid + 1];
      unsigned T = (unsigned)base;
      for (int b = PB - 1; b >= 0; --b) {
        unsigned hb = 0;
        #pragma unroll
        for (int c = 0; c < NSUB; ++c) hb += histp[(base + b) * NSUB + c];
        acc += hb;
        if (acc >= K) { T = (unsigned)(base + b); break; }
      }
      s_thresh = T;
    }
  }
  __syncthreads();

  // =====================================================================
  // Pass 2 (L2-resident; 65.5 MB << 192 MB L2): collect candidates >= T.
  // Composite key = (key<<32) | ~idx  -> descending sort reproduces JAX's
  // stable (value desc, index asc) order.
  // =====================================================================
  {
    const unsigned T = s_thresh;
    const float4* r4 = (const float4*)rowp;
    const int n4 = V >> 2;
    for (int i = tid; i < n4; i += THREADS) {
      if (i + THREADS < n4) __builtin_prefetch(&r4[i + THREADS], 0, 0);
      float4 v = r4[i];
      float vv[4] = {v.x, v.y, v.z, v.w};
      #pragma unroll
      for (int c = 0; c < 4; ++c) {
        unsigned k = fkey(vv[c]);
        if ((k >> 21) >= T) {
          unsigned pos = atomicAdd(&s_ccount, 1u);
          if (pos < CAND_N) {
            unsigned idx = (unsigned)(i * 4 + c);
            cand[pos] = ((unsigned long long)k << 32) |
                        (unsigned long long)(~idx);
          }
        }
      }
    }
    for (int i = (n4 << 2) + tid; i < V; i += THREADS) {
      unsigned k = fkey(rowp[i]);
      if ((k >> 21) >= T) {
        unsigned pos = atomicAdd(&s_ccount, 1u);
        if (pos < CAND_N)
          cand[pos] = ((unsigned long long)k << 32) |
                      (unsigned long long)(~(unsigned)i);
      }
    }
  }
  __syncthreads();

  // ---- bitonic sort (descending) over next-pow2(candidate count) only ----
  {
    unsigned cc = s_ccount; if (cc > CAND_N) cc = CAND_N;
    unsigned N2 = 64; while (N2 < cc) N2 <<= 1;      // uniform across block
    for (unsigned kk = 2; kk <= N2; kk <<= 1) {
      for (unsigned j = kk >> 1; j > 0; j >>= 1) {
        for (unsigned i = (unsigned)tid; i < N2; i += THREADS) {
          unsigned ixj = i ^ j;
          if (ixj > i) {
            unsigned long long a = cand[i], b = cand[ixj];
            bool desc = ((i & kk) == 0);
            if (desc ? (a < b) : (a > b)) { cand[i] = b; cand[ixj] = a; }
          }
        }
        __syncthreads();
      }
    }
  }

  // =====================================================================
  // Finalize in wave 0: exp-weights, WMMA prefix scan, nucleus cut, sample
  // =====================================================================
  {
    int Kc = s_K; Kc = Kc < 1 ? 1 : (Kc > 64 ? 64 : Kc);
    unsigned cc = s_ccount; if (cc > CAND_N) cc = CAND_N;
    const int m0 = (Kc < (int)cc) ? Kc : (int)cc;

    if (tid < 32) {   // whole wave 0 active -> EXEC all ones for WMMA
      float xmax = keyval((unsigned)(cand[0] >> 32)) * TEMP_INV;
      for (int e = tid; e < 256; e += 32) {   // zero-pad full 16x16 W tile
        float w = 0.f; int idx = 0;
        if (e < m0) {
          unsigned long long ce = cand[e];
          float x = keyval((unsigned)(ce >> 32)) * TEMP_INV;
          w = __expf(x - xmax);
          idx = (int)(~(unsigned)(ce & 0xFFFFFFFFull));
        }
        wbuf[e] = w;
        if (e < 64) s_fidx[e] = idx;
      }
      asm volatile("s_wait_dscnt 0x0" ::: "memory");

      // Segmented inclusive scan of wbuf[0..63] as D = L * W:
      //   L = 16x16 lower-triangular ones, W[r][c] = wbuf[16c + r].
      // K=16 accumulated via 4 chained v_wmma_f32_16x16x4_f32.
      const int lane = tid;
      const int half = lane >> 4;
      const int mrow = lane & 15;
      const int ncol = lane & 15;
      v8f acc = {0.f, 0.f, 0.f, 0.f, 0.f, 0.f, 0.f, 0.f};
      #pragma unroll
      for (int kc = 0; kc < 4; ++kc) {
        v2f a, b;
        int ca0 = kc * 4 + half * 2 + 0;    // A: 16x4 f32, K in VGPR dim
        int ca1 = kc * 4 + half * 2 + 1;
        a.x = (ca0 <= mrow) ? 1.f : 0.f;
        a.y = (ca1 <= mrow) ? 1.f : 0.f;
        int rb0 = kc * 4 + (half ? 2 : 0);  // B: 4x16 f32, rows across VGPRs
        int rb1 = kc * 4 + (half ? 3 : 1);
        b.x = wbuf[16 * ncol + rb0];        // unconditional: pad cols are 0
        b.y = wbuf[16 * ncol + rb1];
        acc = __builtin_amdgcn_wmma_f32_16x16x4_f32(
            false, a, false, b, (short)0, acc, false, false);
      }
      if (ncol < 4) {                       // D[i+8*half][ncol] in acc[i]
        #pragma unroll
        for (int i = 0; i < 8; ++i)
          scanbuf[16 * ncol + (i + 8 * half)] = acc[i];
      }
      asm volatile("s_wait_dscnt 0x0" ::: "memory");

      if (tid == 0) {
        int ans = V;
        if (m0 >= 1) {
          float offs[4];
          offs[0] = 0.f;
          for (int c2 = 1; c2 < 4; ++c2)
            offs[c2] = offs[c2 - 1] + scanbuf[16 * (c2 - 1) + 15];
          auto CDF = [&](int e) { return scanbuf[e] + offs[e >> 4]; };

          float Z = CDF(m0 - 1);                  // softmax normalizer (top-k)
          int m = 1;                               // nucleus prefix length
          for (int i2 = 1; i2 < m0; ++i2) {
            if (CDF(i2 - 1) <= TOPP * Z) ++m; else break;
          }
          float Z2 = CDF(m - 1);                  // renormalizer over kept set

          int ki[64]; float kw[64];               // sort kept by vocab index
          for (int i2 = 0; i2 < m; ++i2) { ki[i2] = s_fidx[i2]; kw[i2] = wbuf[i2]; }
          for (int i2 = 1; i2 < m; ++i2) {
            int kii = ki[i2]; float kwi = kw[i2]; int j2 = i2 - 1;
            while (j2 >= 0 && ki[j2] > kii) {
              ki[j2 + 1] = ki[j2]; kw[j2 + 1] = kw[j2]; --j2;
            }
            ki[j2 + 1] = kii; kw[j2 + 1] = kwi;
          }
          float uu = u[row];
          float csum = 0.f;
          for (int i2 = 0; i2 < m; ++i2) {        // inverse-CDF sample
            csum += kw[i2] / Z2;
            if (csum > uu) { ans = ki[i2]; break; }
          }
        } else {
          ans = 0;
        }
        out[row] = ans;
      }
    }
  }
}

extern "C" void kernel_launch(void* const* d_in, const int* in_sizes, int n_in,
                              void* d_out, int out_size, void* d_ws, size_t ws_size,
                              hipStream_t stream) {
  const float* logits = (const float*)d_in[0];
  const float* uvals  = (const float*)d_in[1];
  const int*   topk   = (const int*)d_in[2];
  int B = in_sizes[1];                 // u has B elements
  int V = in_sizes[0] / B;
  int* out = (int*)d_out;
  hipLaunchKernelGGL(sampler_topk_topp_kernel, dim3(B), dim3(THREADS), 0, stream,
                     logits, uvals, topk, out, B, V);
}